// MemoryModule_42623255445545
// MI455X (gfx1250) — compile-verified
//
#include <hip/hip_runtime.h>
#include <math.h>

typedef float v2f __attribute__((ext_vector_type(2)));
typedef float v8f __attribute__((ext_vector_type(8)));

#define B_  8
#define L_  1024
#define D_  1024
#define DH_ 512
#define T_  1023
#define NCHUNK 64
#define ROWS_PER_WG 64
#define WGS_PER_BATCH 8

__device__ __forceinline__ v8f wmma4(v2f a, v2f b, v8f c) {
  // D = A(16x4,f32) * B(4x16,f32) + C(16x16,f32)
  return __builtin_amdgcn_wmma_f32_16x16x4_f32(false, a, false, b, (short)0, c, false, false);
}

// ---------------------------------------------------------------- init
__global__ void init_kernel(float* __restrict__ M, unsigned* __restrict__ bar, size_t nM) {
  size_t i = (size_t)blockIdx.x * blockDim.x + threadIdx.x;
  if (i < nM) M[i] = 0.0f;
  if (i < 64) bar[i] = 0u;
}

// ---------------------------------------------------------------- projections: K = X * W^T  (8192 x 512, K-dim 1024)
// each wave computes a 16x64 strip: one A fragment feeds 4 WMMAs
__global__ void proj_kernel(const float* __restrict__ X,
                            const float* __restrict__ Ws,
                            const float* __restrict__ We,
                            float* __restrict__ KS, float* __restrict__ KE) {
  const float* W = (blockIdx.y == 0) ? Ws : We;
  float* K = (blockIdx.y == 0) ? KS : KE;
  int lane = threadIdx.x & 31;
  int warp = threadIdx.x >> 5;
  int unit = blockIdx.x * 8 + warp;      // 4096 units
  int tm  = unit >> 3;                   // 0..511 row tile
  int tn4 = unit & 7;                    // 0..7  -> 64-wide col strip
  int r16 = lane & 15;
  int kb  = (lane >> 4) << 1;
  int hi  = (lane >> 4) * 8;
  const float* Arow = X + (size_t)(tm * 16 + r16) * D_;
  const float* B0 = W + (size_t)(tn4 * 64 + r16) * D_;   // B[k][n] = W[n*D + k]
  const float* B1 = B0 + (size_t)16 * D_;
  const float* B2 = B0 + (size_t)32 * D_;
  const float* B3 = B0 + (size_t)48 * D_;
  v8f c0 = {0,0,0,0,0,0,0,0}, c1 = c0, c2 = c0, c3 = c0;
  for (int k0 = 0; k0 < D_; k0 += 4) {
    v2f a, b0, b1, b2, b3;
    a.x  = Arow[k0 + kb]; a.y  = Arow[k0 + kb + 1];
    b0.x = B0[k0 + kb];   b0.y = B0[k0 + kb + 1];
    b1.x = B1[k0 + kb];   b1.y = B1[k0 + kb + 1];
    b2.x = B2[k0 + kb];   b2.y = B2[k0 + kb + 1];
    b3.x = B3[k0 + kb];   b3.y = B3[k0 + kb + 1];
    c0 = wmma4(a, b0, c0);
    c1 = wmma4(a, b1, c1);
    c2 = wmma4(a, b2, c2);
    c3 = wmma4(a, b3, c3);
  }
  float* Kout = K + (size_t)(tm * 16 + hi) * DH_ + tn4 * 64 + r16;
#pragma unroll
  for (int r = 0; r < 8; ++r) {
    Kout[(size_t)r * DH_ +  0] = c0[r];
    Kout[(size_t)r * DH_ + 16] = c1[r];
    Kout[(size_t)r * DH_ + 32] = c2[r];
    Kout[(size_t)r * DH_ + 48] = c3[r];
  }
}

// ---------------------------------------------------------------- per-row norms
__global__ void norms_kernel(const float* __restrict__ KS, const float* __restrict__ KE,
                             float* __restrict__ invns, float* __restrict__ invne,
                             float* __restrict__ refs, float* __restrict__ refe) {
  int lane = threadIdx.x & 31;
  int warp = threadIdx.x >> 5;
  int row = blockIdx.x * 8 + warp;       // 0..8191
  const float* ks = KS + (size_t)row * DH_;
  const float* ke = KE + (size_t)row * DH_;
  float ss = 0.f, se = 0.f;
  for (int j = lane; j < DH_; j += 32) {
    float a = ks[j]; ss += a * a;
    float b = ke[j]; se += b * b;
  }
#pragma unroll
  for (int o = 16; o >= 1; o >>= 1) {
    ss += __shfl_xor(ss, o, 32);
    se += __shfl_xor(se, o, 32);
  }
  if (lane == 0) {
    float ns = sqrtf(ss), ne = sqrtf(se);
    invns[row] = 1.0f / fmaxf(ns, 1e-6f);
    invne[row] = 1.0f / fmaxf(ne, 1e-6f);
    refs[row] = 0.7f * ns;
    refe[row] = 0.7f * ne;
  }
}

// ---------------------------------------------------------------- chunked delta-rule scan
// grid (8 rowblocks, 8 batches), 512 threads. M rows [i0,i0+64) owned exclusively.
__global__ void __launch_bounds__(512, 1)
scan_kernel(const float* __restrict__ KS, const float* __restrict__ KE,
            const float* __restrict__ invnsG, const float* __restrict__ invneG,
            const float* __restrict__ refsG, const float* __restrict__ refeG,
            float* __restrict__ Msem, float* __restrict__ Mepi,
            float* __restrict__ V0, unsigned* __restrict__ bar,
            float a_decay) {
  extern __shared__ float smem[];
  float* KsT  = smem;                 // [512][17]
  float* KeT  = KsT + 512 * 17;
  float* Es   = KeT + 512 * 17;       // [512][17]; first 2048 floats reused as Phase-A partial buffer
  float* Ee   = Es  + 512 * 17;
  float* Gs   = Ee  + 512 * 17;       // [16][16] raw Gram
  float* Ge   = Gs + 256;
  float* red1 = Ge + 256;             // [16] warp partials
  float* red2 = red1 + 16;
  float* invns = red2 + 16;           // [16] each below
  float* invne = invns + 16;
  float* refs  = invne + 16;
  float* refe  = refs + 16;
  float* csA   = refe + 16;
  float* ceA   = csA + 16;
  float* usA   = ceA + 16;
  float* ueA   = usA + 16;

  const int tid  = threadIdx.x;
  const int lane = tid & 31;
  const int warp = tid >> 5;
  const int rb = blockIdx.x;
  const int batch = blockIdx.y;
  const int i0 = rb * ROWS_PER_WG;
  const int r16 = lane & 15;
  const int kb  = (lane >> 4) << 1;
  const int hi  = (lane >> 4) * 8;
  const float one_minus_a = 1.0f - a_decay;

  float* Mb_s = Msem + (size_t)batch * DH_ * DH_;
  float* Mb_e = Mepi + (size_t)batch * DH_ * DH_;
  const size_t rowbase = (size_t)batch * L_ * DH_;
  unsigned* mybar = bar + batch;

  for (int c = 0; c < NCHUNK; ++c) {
    // ---- stage chunk K tiles + scalars into LDS (transposed, stride-17 padded), b128 loads
    for (int idx = tid; idx < 16 * 128; idx += 512) {
      int t  = idx >> 7;
      int j4 = (idx & 127) << 2;
      size_t g = rowbase + (size_t)(c * 16 + t) * DH_ + j4;
      float4 vs = *(const float4*)(KS + g);
      float4 ve = *(const float4*)(KE + g);
      KsT[(j4 + 0) * 17 + t] = vs.x; KsT[(j4 + 1) * 17 + t] = vs.y;
      KsT[(j4 + 2) * 17 + t] = vs.z; KsT[(j4 + 3) * 17 + t] = vs.w;
      KeT[(j4 + 0) * 17 + t] = ve.x; KeT[(j4 + 1) * 17 + t] = ve.y;
      KeT[(j4 + 2) * 17 + t] = ve.z; KeT[(j4 + 3) * 17 + t] = ve.w;
    }
    if (tid < 16) {
      int l = batch * L_ + c * 16 + tid;
      invns[tid] = invnsG[l]; invne[tid] = invneG[l];
      refs[tid]  = refsG[l];  refe[tid]  = refeG[l];
    }
    // prefetch next chunk's K rows into cache while Phase A runs
    if (c + 1 < NCHUNK) {
      size_t nb = rowbase + (size_t)((c + 1) * 16 + (tid & 15)) * DH_ + ((size_t)(tid >> 4) << 5);
      __builtin_prefetch(KS + nb, 0, 1);
      __builtin_prefetch(KE + nb, 0, 1);
    }
    __syncthreads();

    const int parity = c & 1;
    float* V0b = V0 + (size_t)(parity * B_ + batch) * 2 * DH_ * 16;

    // ---- Phase A: V0(owned rows) = M * Kn^T, split-K over wave pairs; Gram = K K^T
    {
      int w8   = warp & 7;            // tile: stream/rowtile
      int half = warp >> 3;           // K half
      int st = w8 >> 2, rt = w8 & 3;
      const float* M  = (st == 0) ? Mb_s : Mb_e;
      const float* KT = (st == 0) ? KsT : KeT;
      const float* Mrow = M + (size_t)(i0 + rt * 16 + r16) * DH_;
      v8f acc = {0,0,0,0,0,0,0,0};
      int kbeg = half * 256;
      for (int kk = kbeg; kk < kbeg + 256; kk += 4) {
        v2f a, b;
        a.x = Mrow[kk + kb];            a.y = Mrow[kk + kb + 1];
        b.x = KT[(kk + kb) * 17 + r16]; b.y = KT[(kk + kb + 1) * 17 + r16];
        acc = wmma4(a, b, acc);
      }
      if (half == 1) {                  // stash partial in (currently dead) Es region
        float* P = Es + w8 * 256;
#pragma unroll
        for (int r = 0; r < 8; ++r) P[(hi + r) * 16 + r16] = acc[r];
      }
      __syncthreads();
      if (half == 0) {                  // combine, scale by invn, store V0
        const float* P = Es + w8 * 256;
        const float* inv = (st == 0) ? invns : invne;
        float sc = inv[r16];
        float* dst = V0b + (size_t)st * DH_ * 16;
#pragma unroll
        for (int r = 0; r < 8; ++r)
          dst[(size_t)(i0 + rt * 16 + hi + r) * 16 + r16] =
              (acc[r] + P[(hi + r) * 16 + r16]) * sc;
      } else if (w8 < 2) {              // two freed waves compute the Gram matrices
        const float* KT2 = (w8 == 0) ? KsT : KeT;
        float* G = (w8 == 0) ? Gs : Ge;
        v8f g2 = {0,0,0,0,0,0,0,0};
        for (int kk = 0; kk < DH_; kk += 4) {
          v2f a;   // A and B fragments coincide for K K^T
          a.x = KT2[(kk + kb) * 17 + r16]; a.y = KT2[(kk + kb + 1) * 17 + r16];
          g2 = wmma4(a, a, g2);
        }
#pragma unroll
        for (int r = 0; r < 8; ++r) G[(hi + r) * 16 + r16] = g2[r];
      }
    }

    // ---- release V0, then batch-wide (8-WG) barrier + acquire
    __threadfence();
    __syncthreads();
    if (tid == 0) {
      __hip_atomic_fetch_add(mybar, 1u, __ATOMIC_RELEASE, __HIP_MEMORY_SCOPE_AGENT);
      unsigned target = (unsigned)(WGS_PER_BATCH * (c + 1));
      while (__hip_atomic_load(mybar, __ATOMIC_ACQUIRE, __HIP_MEMORY_SCOPE_AGENT) < target)
        __builtin_amdgcn_s_sleep(1);
    }
    __syncthreads();
    __builtin_amdgcn_fence(__ATOMIC_ACQUIRE, "agent");

    // ---- Phase B: sequential in-chunk recursion; thread i owns component i (full 512-vec, redundant per WG)
    {
      const int i = tid;
      const float* V0s = V0b;
      const float* V0e = V0b + (size_t)DH_ * 16;
      for (int t = 0; t < 16; ++t) {
        float es = KsT[i * 17 + t] - V0s[(size_t)i * 16 + t];
        float ee = KeT[i * 17 + t] - V0e[(size_t)i * 16 + t];
        for (int s = 0; s < t; ++s) {
          float cofs = csA[s] * invns[s] * invns[t] * Gs[s * 16 + t];
          float cofe = ceA[s] * invne[s] * invne[t] * Ge[s * 16 + t];
          es -= cofs * Es[i * 17 + s];
          ee -= cofe * Ee[i * 17 + s];
        }
        Es[i * 17 + t] = es;
        Ee[i * 17 + t] = ee;
        float ps = es * es, pe = ee * ee;
#pragma unroll
        for (int o = 16; o >= 1; o >>= 1) {
          ps += __shfl_xor(ps, o, 32);
          pe += __shfl_xor(pe, o, 32);
        }
        if (lane == 0) { red1[warp] = ps; red2[warp] = pe; }
        __syncthreads();
        if (tid == 0) {
          float ss = 0.f, se = 0.f;
#pragma unroll
          for (int w = 0; w < 16; ++w) { ss += red1[w]; se += red2[w]; }
          int tg = c * 16 + t;
          float errs = sqrtf(ss);
          float erre = sqrtf(se);
          float fire = ((tg < T_) && (errs >= refs[t] || erre >= refe[t])) ? 1.0f : 0.0f;
          float cs = one_minus_a * fire;
          float ce = cs * ((float)(tg + 1) / (float)L_);
          csA[t] = cs; ceA[t] = ce;
          usA[t] = cs * invns[t];   // fold kn normalization into update coeff
          ueA[t] = ce * invne[t];
        }
        __syncthreads();
      }
    }

    // ---- Phase C: M(owned rows) += (E .* u) * K   (rank-16, WMMA), all 16 waves
    for (int tj = warp; tj < 256; tj += 16) {
      int st  = tj >> 7;
      int rem = tj & 127;
      int rt  = rem >> 5;
      int ct  = rem & 31;
      float* M        = (st == 0) ? Mb_s : Mb_e;
      const float* E  = (st == 0) ? Es : Ee;
      const float* KT = (st == 0) ? KsT : KeT;
      const float* U  = (st == 0) ? usA : ueA;
      float* Mt = M + (size_t)(i0 + rt * 16) * DH_ + ct * 16;
      v8f acc;
#pragma unroll
      for (int r = 0; r < 8; ++r) acc[r] = Mt[(size_t)(hi + r) * DH_ + r16];
      const float* Erow = E  + (size_t)(i0 + rt * 16 + r16) * 17;
      const float* Kcol = KT + (size_t)(ct * 16 + r16) * 17;
#pragma unroll
      for (int kk = 0; kk < 16; kk += 4) {
        v2f a, b;
        a.x = Erow[kk + kb] * U[kk + kb];
        a.y = Erow[kk + kb + 1] * U[kk + kb + 1];
        b.x = Kcol[kk + kb];
        b.y = Kcol[kk + kb + 1];
        acc = wmma4(a, b, acc);
      }
#pragma unroll
      for (int r = 0; r < 8; ++r) Mt[(size_t)(hi + r) * DH_ + r16] = acc[r];
    }
    __threadfence();
    __syncthreads();
  }
}

// ---------------------------------------------------------------- readout: gate, matvecs, output proj, layernorm
__global__ void final_kernel(const float* __restrict__ X,
                             const float* __restrict__ Wg, const float* __restrict__ bg,
                             const float* __restrict__ Wo, const float* __restrict__ bo,
                             const float* __restrict__ gamma, const float* __restrict__ beta,
                             const float* __restrict__ KS, const float* __restrict__ KE,
                             const float* __restrict__ invnsG, const float* __restrict__ invneG,
                             const float* __restrict__ Msem, const float* __restrict__ Mepi,
                             float* __restrict__ out) {
  __shared__ float qn[DH_ * 2];
  __shared__ float rr[D_];
  __shared__ float oo[D_];
  __shared__ float red[512];
  __shared__ float gshared, mu_s, var_s;
  int b = blockIdx.x;
  int tid = threadIdx.x;
  int qrow = b * L_ + (L_ - 1);
  if (tid < DH_) {
    qn[tid]        = KS[(size_t)qrow * DH_ + tid] * invnsG[qrow];
    qn[DH_ + tid]  = KE[(size_t)qrow * DH_ + tid] * invneG[qrow];
  }
  const float* q = X + (size_t)qrow * D_;
  float acc = 0.f;
  for (int d = tid; d < D_; d += 512) acc += q[d] * Wg[d];
  red[tid] = acc; __syncthreads();
  for (int off = 256; off >= 1; off >>= 1) { if (tid < off) red[tid] += red[tid + off]; __syncthreads(); }
  if (tid == 0) gshared = 1.0f / (1.0f + expf(-(red[0] + bg[0])));
  __syncthreads();
  float g = gshared;
  if (tid < DH_) {
    const float* Ms = Msem + ((size_t)b * DH_ + tid) * DH_;
    const float* Me = Mepi + ((size_t)b * DH_ + tid) * DH_;
    float rs = 0.f, re = 0.f;
    for (int j = 0; j < DH_; ++j) { rs += Ms[j] * qn[j]; re += Me[j] * qn[DH_ + j]; }
    rr[tid] = g * rs;
    rr[DH_ + tid] = g * re;
  }
  __syncthreads();
  for (int od = tid; od < D_; od += 512) {
    const float* w = Wo + (size_t)od * D_;
    float s = 0.f;
    for (int d = 0; d < D_; ++d) s += rr[d] * w[d];
    oo[od] = s + bo[od];
  }
  __syncthreads();
  float m = 0.f;
  for (int od = tid; od < D_; od += 512) m += oo[od];
  red[tid] = m; __syncthreads();
  for (int off = 256; off >= 1; off >>= 1) { if (tid < off) red[tid] += red[tid + off]; __syncthreads(); }
  if (tid == 0) mu_s = red[0] / (float)D_;
  __syncthreads();
  float mu = mu_s, v = 0.f;
  for (int od = tid; od < D_; od += 512) { float d0 = oo[od] - mu; v += d0 * d0; }
  red[tid] = v; __syncthreads();
  for (int off = 256; off >= 1; off >>= 1) { if (tid < off) red[tid] += red[tid + off]; __syncthreads(); }
  if (tid == 0) var_s = red[0] / (float)D_;
  __syncthreads();
  float rstd = rsqrtf(var_s + 1e-5f);
  for (int od = tid; od < D_; od += 512)
    out[(size_t)b * D_ + od] = (oo[od] - mu) * rstd * gamma[od] + beta[od];
}

// ---------------------------------------------------------------- launch
extern "C" void kernel_launch(void* const* d_in, const int* in_sizes, int n_in,
                              void* d_out, int out_size, void* d_ws, size_t ws_size,
                              hipStream_t stream) {
  (void)in_sizes; (void)n_in; (void)out_size; (void)ws_size;
  const float* X  = (const float*)d_in[0];
  const float* Ws = (const float*)d_in[1];
  const float* We = (const float*)d_in[2];
  const float* Wg = (const float*)d_in[3];
  const float* bg = (const float*)d_in[4];
  const float* Wo = (const float*)d_in[5];
  const float* bo = (const float*)d_in[6];
  const float* gamma = (const float*)d_in[7];
  const float* beta  = (const float*)d_in[8];
  float* out = (float*)d_out;
  float* ws = (float*)d_ws;

  const size_t nK = (size_t)B_ * L_ * DH_;
  float* KS    = ws;
  float* KE    = KS + nK;
  float* invns = KE + nK;
  float* invne = invns + B_ * L_;
  float* refs  = invne + B_ * L_;
  float* refe  = refs + B_ * L_;
  float* Msem  = refe + B_ * L_;
  float* Mepi  = Msem + (size_t)B_ * DH_ * DH_;
  float* V0    = Mepi + (size_t)B_ * DH_ * DH_;
  unsigned* bar = (unsigned*)(V0 + (size_t)2 * B_ * 2 * DH_ * 16);

  const float a_decay = powf(0.95f, 96.0f / 1024.0f);
  const size_t nM = (size_t)2 * B_ * DH_ * DH_;

  init_kernel<<<dim3((unsigned)((nM + 255) / 256)), dim3(256), 0, stream>>>(Msem, bar, nM);
  proj_kernel<<<dim3(512, 2), dim3(256), 0, stream>>>(X, Ws, We, KS, KE);
  norms_kernel<<<dim3(1024), dim3(256), 0, stream>>>(KS, KE, invns, invne, refs, refe);
  size_t smem = (size_t)(4 * 512 * 17 + 2 * 256 + 2 * 16 + 8 * 16) * sizeof(float);
  scan_kernel<<<dim3(8, 8), dim3(512), smem, stream>>>(KS, KE, invns, invne, refs, refe,
                                                       Msem, Mepi, V0, bar, a_decay);
  final_kernel<<<dim3(8), dim3(512), 0, stream>>>(X, Wg, bg, Wo, bo, gamma, beta,
                                                  KS, KE, invns, invne, Msem, Mepi, out);
}